// _ScaledDotProductAttention_22634477650274
// MI455X (gfx1250) — compile-verified
//
#include <hip/hip_runtime.h>
#include <math.h>

typedef __attribute__((ext_vector_type(2))) float v2f;
typedef __attribute__((ext_vector_type(8))) float v8f;

constexpr int SEQ = 2048;
constexpr int HD  = 64;
constexpr int NBH = 32;           // B*H = 2*16
constexpr float QK_SCALE = 0.125f; // 64^-0.5

// One workgroup = 256 threads = 8 waves handles 16 query rows x full SEQ.
// Wave w owns score/weight columns [w*256, w*256+256).
__launch_bounds__(256, 1)
__global__ void fused_attn_kernel(const float* __restrict__ q,
                                  const float* __restrict__ k,
                                  const float* __restrict__ v,
                                  const float* __restrict__ prev,
                                  float* __restrict__ out_o,
                                  float* __restrict__ out_w,
                                  float* __restrict__ out_s)
{
    __shared__ float redmax[16][8];
    __shared__ float redsum[16][8];
    __shared__ float wtile[8][16][17];   // per-wave C/D->A layout staging
    __shared__ float pout[8][16 * 64];   // per-wave partial output tiles

    const int tid  = threadIdx.x;
    const int w    = tid >> 5;       // wave id 0..7
    const int lane = tid & 31;
    const int M    = lane & 15;      // N-index / M-index within WMMA frags
    const int half = lane >> 4;      // 0: lanes 0-15, 1: lanes 16-31

    const int blk   = blockIdx.x;
    const int bh    = blk >> 7;           // / (SEQ/16 = 128)
    const int qbase = (blk & 127) << 4;   // 16-row block within sequence

    const float* qp = q    + (size_t)bh * SEQ * HD;
    const float* kp = k    + (size_t)bh * HD  * SEQ;   // k is [d, s]
    const float* vp = v    + (size_t)bh * SEQ * HD;
    const float* pp = prev + (size_t)bh * SEQ * SEQ;
    float* sp = out_s + (size_t)bh * SEQ * SEQ;
    float* wp = out_w + (size_t)bh * SEQ * SEQ;
    float* op = out_o + (size_t)bh * SEQ * HD;

    // Prefetch the prev block (16 rows x 2048 cols = 128KB) into cache while
    // GEMM1 runs: emits global_prefetch_b8, 128B per lane-slot.
    {
        const float* base = pp + (size_t)qbase * SEQ;
        #pragma unroll
        for (int i = 0; i < 4; ++i)
            __builtin_prefetch(base + (size_t)(i * 256 + tid) * 32, 0, 3);
    }

    // ---------------- GEMM1: scores = Q[16xD] @ K[DxS] (this wave: 256 cols)
    v8f acc[16] = {};
    const int colB = w << 8; // w*256

    for (int t = 0; t < 16; ++t) {
        const int kk = 4 * t + 2 * half;
        // A frag: lane holds q[row=M][kk], q[row=M][kk+1]
        v2f a = *(const v2f*)(qp + (size_t)(qbase + M) * HD + kk);
        #pragma unroll
        for (int c = 0; c < 16; ++c) {
            const int col = colB + c * 16 + M;
            v2f b;
            b.x = kp[(size_t)kk       * SEQ + col];
            b.y = kp[(size_t)(kk + 1) * SEQ + col];
            acc[c] = __builtin_amdgcn_wmma_f32_16x16x4_f32(
                false, a, false, b, (short)0, acc[c], false, false);
        }
    }

    // ---------------- epilogue: scale + prev, write scores
    #pragma unroll
    for (int c = 0; c < 16; ++c) {
        const int col = colB + c * 16 + M;
        #pragma unroll
        for (int r = 0; r < 8; ++r) {
            const int row = qbase + r + 8 * half;
            float sc = acc[c][r] * QK_SCALE + pp[(size_t)row * SEQ + col];
            acc[c][r] = sc;
            sp[(size_t)row * SEQ + col] = sc;
        }
    }

    // ---------------- softmax across full rows
    float rmax[8];
    #pragma unroll
    for (int r = 0; r < 8; ++r) {
        float m = -3.0e38f;
        #pragma unroll
        for (int c = 0; c < 16; ++c) m = fmaxf(m, acc[c][r]);
        #pragma unroll
        for (int mk = 1; mk < 16; mk <<= 1)      // stays within 16-lane half
            m = fmaxf(m, __shfl_xor(m, mk, 32));
        rmax[r] = m;
        if (M == 0) redmax[r + 8 * half][w] = m;
    }
    __syncthreads();
    #pragma unroll
    for (int r = 0; r < 8; ++r) {
        float m = rmax[r];
        #pragma unroll
        for (int ww = 0; ww < 8; ++ww) m = fmaxf(m, redmax[r + 8 * half][ww]);
        rmax[r] = m;
    }

    float rinv[8];
    #pragma unroll
    for (int r = 0; r < 8; ++r) {
        float s = 0.0f;
        #pragma unroll
        for (int c = 0; c < 16; ++c) {
            float e = __expf(acc[c][r] - rmax[r]);
            acc[c][r] = e;
            s += e;
        }
        #pragma unroll
        for (int mk = 1; mk < 16; mk <<= 1)
            s += __shfl_xor(s, mk, 32);
        rinv[r] = s;
        if (M == 0) redsum[r + 8 * half][w] = s;
    }
    __syncthreads();
    #pragma unroll
    for (int r = 0; r < 8; ++r) {
        float s = 0.0f;
        #pragma unroll
        for (int ww = 0; ww < 8; ++ww) s += redsum[r + 8 * half][ww];
        rinv[r] = 1.0f / s;
    }

    // weights = exp * inv_sum; write
    #pragma unroll
    for (int c = 0; c < 16; ++c) {
        const int col = colB + c * 16 + M;
        #pragma unroll
        for (int r = 0; r < 8; ++r) {
            const int row = qbase + r + 8 * half;
            float wv = acc[c][r] * rinv[r];
            acc[c][r] = wv;
            wp[(size_t)row * SEQ + col] = wv;
        }
    }

    // ---------------- GEMM2: partial O += W_chunk @ V_chunk over this wave's
    // 256-col K slice. Weights sit in C/D layout; stage each 16x16 tile
    // through per-wave LDS to re-read in A layout (DS ops in-order per wave).
    v8f oacc[4] = {};
    #pragma unroll
    for (int c = 0; c < 16; ++c) {
        const int KG = colB + c * 16;
        #pragma unroll
        for (int r = 0; r < 8; ++r)
            wtile[w][r + 8 * half][M] = acc[c][r];
        #pragma unroll
        for (int t2 = 0; t2 < 4; ++t2) {
            const int kk2 = 4 * t2 + 2 * half;
            v2f a;
            a.x = wtile[w][M][kk2];
            a.y = wtile[w][M][kk2 + 1];
            #pragma unroll
            for (int ct = 0; ct < 4; ++ct) {
                v2f b;
                b.x = vp[(size_t)(KG + kk2)     * HD + ct * 16 + M];
                b.y = vp[(size_t)(KG + kk2 + 1) * HD + ct * 16 + M];
                oacc[ct] = __builtin_amdgcn_wmma_f32_16x16x4_f32(
                    false, a, false, b, (short)0, oacc[ct], false, false);
            }
        }
    }

    // ---------------- deterministic cross-wave reduction of output partials
    #pragma unroll
    for (int ct = 0; ct < 4; ++ct)
        #pragma unroll
        for (int r = 0; r < 8; ++r)
            pout[w][(r + 8 * half) * 64 + ct * 16 + M] = oacc[ct][r];
    __syncthreads();
    #pragma unroll
    for (int i = 0; i < 4; ++i) {
        const int e = i * 256 + tid;   // 16*64 = 1024 outputs
        float s = 0.0f;
        #pragma unroll
        for (int ww = 0; ww < 8; ++ww) s += pout[ww][e];
        op[(size_t)(qbase + (e >> 6)) * HD + (e & 63)] = s;
    }
}

extern "C" void kernel_launch(void* const* d_in, const int* in_sizes, int n_in,
                              void* d_out, int out_size, void* d_ws, size_t ws_size,
                              hipStream_t stream) {
    const float* q    = (const float*)d_in[0];
    const float* k    = (const float*)d_in[1];
    const float* v    = (const float*)d_in[2];
    const float* prev = (const float*)d_in[3];

    float* out   = (float*)d_out;
    float* out_o = out;                                   // [BH, S, D]
    float* out_w = out_o + (size_t)NBH * SEQ * HD;        // [BH, S, S]
    float* out_s = out_w + (size_t)NBH * SEQ * SEQ;       // [BH, S, S]

    dim3 grid(NBH * (SEQ / 16));   // 4096 workgroups
    fused_attn_kernel<<<grid, 256, 0, stream>>>(q, k, v, prev, out_o, out_w, out_s);
}